// PNASimpleLayer_48533130444874
// MI455X (gfx1250) — compile-verified
//
#include <hip/hip_runtime.h>
#include <math.h>

#define F_DIM 64
#define AVG_D_LOG 2.833f
#define EPS_STD 1e-5f
#define EPS_BN 1e-5f

typedef __attribute__((ext_vector_type(2))) float v2f;
typedef __attribute__((ext_vector_type(8))) float v8f;

// ---- monotonic uint encoding for float atomic max/min via u32 atomics ----
__device__ __forceinline__ unsigned int fenc(float x) {
    unsigned int u = __float_as_uint(x);
    return (u & 0x80000000u) ? ~u : (u | 0x80000000u);
}
__device__ __forceinline__ float fdec(unsigned int k) {
    return (k & 0x80000000u) ? __uint_as_float(k ^ 0x80000000u)
                             : __uint_as_float(~k);
}

// ---------------- Stage 0: init workspace ----------------
__global__ void pna_init(float* s, float* s2, unsigned int* mxu, unsigned int* mnu,
                         float* deg, float* colsum, float* colsumsq,
                         int nTot, int nNodes) {
    int i = blockIdx.x * blockDim.x + threadIdx.x;
    if (i < nTot) {
        s[i] = 0.0f;
        s2[i] = 0.0f;
        mxu[i] = 0u;          // encodes "-infinity-most" key
        mnu[i] = 0xFFFFFFFFu; // encodes "+infinity-most" key
    }
    if (i < nNodes) deg[i] = 0.0f;
    if (i < F_DIM) { colsum[i] = 0.0f; colsumsq[i] = 0.0f; }
}

// ---------------- Stage 1: edge scatter-aggregation ----------------
// one thread per (edge, feature)
__global__ void pna_edge(const float* __restrict__ h,
                         const int* __restrict__ src, const int* __restrict__ dst,
                         float* s, float* s2, unsigned int* mxu, unsigned int* mnu,
                         float* deg, int nEdges) {
    long long tid = (long long)blockIdx.x * blockDim.x + threadIdx.x;
    int e = (int)(tid >> 6);
    int f = (int)(tid & 63);
    if (e >= nEdges) return;
    int si = src[e];
    int di = dst[e];
    float m = h[si * F_DIM + f];
    int base = di * F_DIM + f;
    unsafeAtomicAdd(&s[base], m);
    unsafeAtomicAdd(&s2[base], m * m);
    unsigned int k = fenc(m);
    atomicMax(&mxu[base], k);
    atomicMin(&mnu[base], k);
    if (f == 0) unsafeAtomicAdd(&deg[di], 1.0f);
}

// ---------------- Stage 2: per-node finalize (in place) ----------------
// s -> mean, s2 -> std, mxu -> max(float), mnu -> min(float); also amp/att scalars
__global__ void pna_finalize(float* s, float* s2, unsigned int* mxu, unsigned int* mnu,
                             const float* __restrict__ deg,
                             float* ampS, float* attS, int nTot) {
    int i = blockIdx.x * blockDim.x + threadIdx.x;
    if (i >= nTot) return;
    int n = i >> 6;
    int f = i & 63;
    float d = deg[n];
    bool has = d > 0.0f;
    float degc = fmaxf(d, 1.0f);
    float mean = s[i] / degc;
    float var = fmaxf(s2[i] / degc - mean * mean, 0.0f);
    float sd = sqrtf(var + EPS_STD);
    float mx = fdec(mxu[i]);
    float mn = fdec(mnu[i]);
    if (!has) { mean = 0.0f; sd = 0.0f; mx = 0.0f; mn = 0.0f; }
    s[i] = mean;
    s2[i] = sd;
    ((float*)mxu)[i] = mx;
    ((float*)mnu)[i] = mn;
    if (f == 0) {
        float logD = logf(d + 1.0f);
        ampS[n] = logD / AVG_D_LOG;
        attS[n] = has ? (AVG_D_LOG / fmaxf(logD, 1e-12f)) : 0.0f;
    }
}

// ---------------- Stage 3: fused scaler GEMM with f32 WMMA ----------------
// y[N,64] = (agg @ W_id) + a_row*(agg @ W_amp) + t_row*(agg @ W_att) + bias
// agg columns: k in [0,64)->mean, [64,128)->max, [128,192)->min, [192,256)->std
// One wave computes a 16x64 output tile; W staged via LDS in 16-row k-chunks.
// LDS holds W pair-swizzled: Wl[s][k>>1][col*2 + (k&1)] so that a B fragment
// (W[k][col], W[k+1][col]) is one aligned 8-byte ds_load_b64 into an even
// VGPR pair -> feeds v_wmma SRC1 directly, no register re-pairing movs.
__global__ void __launch_bounds__(256)
pna_gemm(const float* __restrict__ Amean, const float* __restrict__ Amax,
         const float* __restrict__ Amin, const float* __restrict__ Astd,
         const float* __restrict__ W, const float* __restrict__ bias,
         const float* __restrict__ ampS, const float* __restrict__ attS,
         float* __restrict__ y, int nNodes) {
    __shared__ float Wl[3 * 16 * F_DIM]; // 12 KB, pair-swizzled (see above)

    const int lane = threadIdx.x & 31;
    const int wave = threadIdx.x >> 5;
    const int tile = blockIdx.x * 8 + wave;
    const int row0 = tile * 16;
    const bool valid = (row0 + 15) < nNodes;

    v8f acc[3][4];
#pragma unroll
    for (int sIdx = 0; sIdx < 3; ++sIdx)
#pragma unroll
        for (int nt = 0; nt < 4; ++nt) acc[sIdx][nt] = (v8f){};

    // A-fragment addressing (ISA: A 16x4 f32: lanes 0-15 K0/K1, lanes 16-31 K2/K3)
    const int arow = row0 + (lane & 15);
    const int klaneoff = (lane >> 4) << 1; // 0 or 2 (always even)

    const v2f* Wp = (const v2f*)Wl; // pair view: Wp[s*512 + (k>>1)*64 + col]

    for (int kc = 0; kc < 16; ++kc) { // 16 chunks of K=16  (total K=256)
        // cooperative load of W chunk with pair-swizzle:
        // global element (sIdx*256 + kc*16 + k)*64 + n
        //   -> Wl[sIdx*1024 + (k>>1)*128 + n*2 + (k&1)]
        for (int i = threadIdx.x; i < 3 * 16 * F_DIM; i += 256) {
            int sIdx = i >> 10;
            int r = i & 1023;
            int k = r >> 6;
            int n = r & 63;
            Wl[sIdx * 1024 + (k >> 1) * 128 + n * 2 + (k & 1)] =
                W[(sIdx * 256 + kc * 16) * F_DIM + r];
        }
        __syncthreads();

        if (valid) {
            // all k in this chunk live in the same 64-wide agg sub-block
            const float* Aarr;
            switch (kc >> 2) {
                case 0: Aarr = Amean; break;
                case 1: Aarr = Amax; break;
                case 2: Aarr = Amin; break;
                default: Aarr = Astd; break;
            }
#pragma unroll
            for (int k4 = 0; k4 < 4; ++k4) {
                const int kk = kc * 16 + k4 * 4;       // global k base
                const int f = (kk + klaneoff) & 63;    // even
                v2f a = *(const v2f*)&Aarr[arow * F_DIM + f]; // global_load_b64

                const int kp = k4 * 2 + (lane >> 4);   // (k_local >> 1)
#pragma unroll
                for (int sIdx = 0; sIdx < 3; ++sIdx) {
#pragma unroll
                    for (int nt = 0; nt < 4; ++nt) {
                        const int col = nt * 16 + (lane & 15);
                        v2f b = Wp[sIdx * 512 + kp * 64 + col]; // ds_load_b64
                        acc[sIdx][nt] = __builtin_amdgcn_wmma_f32_16x16x4_f32(
                            false, a, false, b, (short)0, acc[sIdx][nt],
                            false, false);
                    }
                }
            }
        }
        __syncthreads();
    }

    if (valid) {
        // C/D layout: VGPR j -> row j (lanes 0-15) / row j+8 (lanes 16-31), col = lane&15
#pragma unroll
        for (int j = 0; j < 8; ++j) {
            const int row = row0 + j + ((lane >> 4) << 3);
            const float av = ampS[row];
            const float tv = attS[row];
#pragma unroll
            for (int nt = 0; nt < 4; ++nt) {
                const int col = nt * 16 + (lane & 15);
                float v = acc[0][nt][j] + av * acc[1][nt][j] + tv * acc[2][nt][j]
                          + bias[col];
                y[row * F_DIM + col] = v;
            }
        }
    }
}

// ---------------- Stage 4: batch-norm column statistics ----------------
__global__ void pna_colstats(const float* __restrict__ y,
                             float* colsum, float* colsumsq, int nNodes) {
    __shared__ float ls[256], ls2[256];
    const int col = threadIdx.x & 63;
    const int grp = threadIdx.x >> 6; // 0..3
    float s = 0.0f, s2 = 0.0f;
    for (int row = blockIdx.x * 4 + grp; row < nNodes; row += gridDim.x * 4) {
        float v = y[row * F_DIM + col];
        s += v;
        s2 += v * v;
    }
    ls[threadIdx.x] = s;
    ls2[threadIdx.x] = s2;
    __syncthreads();
    if (grp == 0) {
        s = ls[col] + ls[col + 64] + ls[col + 128] + ls[col + 192];
        s2 = ls2[col] + ls2[col + 64] + ls2[col + 128] + ls2[col + 192];
        unsafeAtomicAdd(&colsum[col], s);
        unsafeAtomicAdd(&colsumsq[col], s2);
    }
}

// ---------------- Stage 5: BN apply + ReLU + residual ----------------
__global__ void pna_bn_apply(const float* __restrict__ y, const float* __restrict__ h,
                             const float* __restrict__ gamma, const float* __restrict__ beta,
                             const float* __restrict__ colsum, const float* __restrict__ colsumsq,
                             float* __restrict__ out, int nTot, float invN) {
    int i = blockIdx.x * blockDim.x + threadIdx.x;
    if (i >= nTot) return;
    int f = i & 63;
    float mu = colsum[f] * invN;
    float var = colsumsq[f] * invN - mu * mu;
    float xn = gamma[f] * (y[i] - mu) * rsqrtf(var + EPS_BN) + beta[f];
    out[i] = h[i] + fmaxf(xn, 0.0f);
}

// ---------------- host launcher ----------------
extern "C" void kernel_launch(void* const* d_in, const int* in_sizes, int n_in,
                              void* d_out, int out_size, void* d_ws, size_t ws_size,
                              hipStream_t stream) {
    const float* h     = (const float*)d_in[0];
    const int*   src   = (const int*)d_in[1];
    const int*   dst   = (const int*)d_in[2];
    const float* W     = (const float*)d_in[3];
    const float* bias  = (const float*)d_in[4];
    const float* gamma = (const float*)d_in[5];
    const float* beta  = (const float*)d_in[6];
    float* out = (float*)d_out;

    const int nNodes = in_sizes[0] / F_DIM;
    const int nEdges = in_sizes[1];
    const int nTot = nNodes * F_DIM;

    // carve workspace (256B aligned blocks)
    auto align256 = [](size_t x) { return (x + 255) & ~(size_t)255; };
    char* p = (char*)d_ws;
    float* deg  = (float*)p;        p += align256((size_t)nNodes * 4);
    float* s    = (float*)p;        p += align256((size_t)nTot * 4);
    float* s2   = (float*)p;        p += align256((size_t)nTot * 4);
    unsigned int* mxu = (unsigned int*)p; p += align256((size_t)nTot * 4);
    unsigned int* mnu = (unsigned int*)p; p += align256((size_t)nTot * 4);
    float* ampS = (float*)p;        p += align256((size_t)nNodes * 4);
    float* attS = (float*)p;        p += align256((size_t)nNodes * 4);
    float* y    = (float*)p;        p += align256((size_t)nTot * 4);
    float* colsum   = (float*)p;    p += 256;
    float* colsumsq = (float*)p;    p += 256;

    const int T = 256;

    // Stage 0: init
    pna_init<<<(nTot + T - 1) / T, T, 0, stream>>>(s, s2, mxu, mnu, deg,
                                                   colsum, colsumsq, nTot, nNodes);
    // Stage 1: edge aggregation (one thread per edge-feature)
    long long eThreads = (long long)nEdges * F_DIM;
    int eBlocks = (int)((eThreads + T - 1) / T);
    pna_edge<<<eBlocks, T, 0, stream>>>(h, src, dst, s, s2, mxu, mnu, deg, nEdges);
    // Stage 2: finalize aggregates
    pna_finalize<<<(nTot + T - 1) / T, T, 0, stream>>>(s, s2, mxu, mnu, deg,
                                                       ampS, attS, nTot);
    // Stage 3: fused WMMA GEMM (16-row tile per wave, 8 waves per block)
    int tiles = (nNodes + 15) / 16;
    int gBlocks = (tiles + 7) / 8;
    pna_gemm<<<gBlocks, 256, 0, stream>>>( s, (float*)mxu, (float*)mnu, s2,
                                           W, bias, ampS, attS, y, nNodes);
    // Stage 4: column stats
    pna_colstats<<<256, 256, 0, stream>>>(y, colsum, colsumsq, nNodes);
    // Stage 5: BN + ReLU + residual
    pna_bn_apply<<<(nTot + T - 1) / T, T, 0, stream>>>(y, h, gamma, beta,
                                                       colsum, colsumsq, out,
                                                       nTot, 1.0f / (float)nNodes);
}